// TripleLoss_90967407329287
// MI455X (gfx1250) — compile-verified
//
#include <hip/hip_runtime.h>

// Problem constants (from the reference)
#define NROWS 4096
#define DIM   1024
#define TILE  128            // C-tile per workgroup (128x128)
#define NTIL  (NROWS / TILE) // 32 tiles per dimension
#define KT    16             // K slice staged in LDS per pipeline stage
#define LSTR  20             // padded LDS row stride in floats (KT + 4) -> 80B rows

typedef __attribute__((ext_vector_type(2))) float v2f;
typedef __attribute__((ext_vector_type(8))) float v8f;

// -------------------------------------------------------------------------
// Phase 1: fused  dist-tile = X[I:I+128,:] @ X[J:J+128,:]^T  (exact fp32 WMMA)
//          with double-buffered GLOBAL_LOAD_ASYNC_TO_LDS_B128 staging,
//          + masked per-row argmin (hardest-positive / easiest-negative)
//          -> per (row, col-tile) partial {ap_val, ap_idx, an_val, an_idx}
// -------------------------------------------------------------------------
__global__ __launch_bounds__(256) void triplet_phase1(
    const float* __restrict__ X, const int* __restrict__ labels,
    float4* __restrict__ part)
{
  __shared__ float As[2][TILE * LSTR];
  __shared__ float Bs[2][TILE * LSTR];
  __shared__ int   labR[TILE];
  __shared__ int   labC[TILE];
  __shared__ float rvAP[2][TILE];
  __shared__ int   riAP[2][TILE];
  __shared__ float rvAN[2][TILE];
  __shared__ int   riAN[2][TILE];

  const int jt = blockIdx.x;        // column tile
  const int it = blockIdx.y;        // row tile
  const int I  = it * TILE;
  const int J  = jt * TILE;

  const int tid  = threadIdx.x;
  const int wid  = tid >> 5;        // wave id 0..7 (wave32)
  const int lane = tid & 31;
  const int wrow = wid & 3;         // 0..3  -> 32-row band
  const int wcol = wid >> 2;        // 0..1  -> 64-col band
  const int lhi  = lane >> 4;       // 0/1   (K-pair select per ISA A/B layout)
  const int llo  = lane & 15;       // M / N within 16

  if (tid < TILE) {
    labR[tid] = labels[I + tid];
    labC[tid] = labels[J + tid];
  }

  // Async stage of one K-slice: 128 rows x 4 float4 per row per tile,
  // 2 float4 per thread per tile -> 4 async b128 per thread (ASYNCcnt += 4).
  // GVS addressing: SGPR64 base (X) + VGPR32 byte offset; VDST = LDS address.
#define STAGE(KB, BUF)                                                         \
  do {                                                                         \
    _Pragma("unroll")                                                          \
    for (int i = 0; i < 2; ++i) {                                              \
      int q   = tid + (i << 8);                                                \
      int row = q >> 2;                                                        \
      int c4  = q & 3;                                                         \
      unsigned goffA = (unsigned)((((I + row) * DIM) + (KB) + (c4 << 2)) * 4); \
      unsigned goffB = (unsigned)((((J + row) * DIM) + (KB) + (c4 << 2)) * 4); \
      unsigned ldsA  = (unsigned)(uintptr_t)&As[BUF][row * LSTR + (c4 << 2)];  \
      unsigned ldsB  = (unsigned)(uintptr_t)&Bs[BUF][row * LSTR + (c4 << 2)];  \
      asm volatile("global_load_async_to_lds_b128 %0, %1, %2"                  \
                   :: "v"(ldsA), "v"(goffA), "s"(X) : "memory");               \
      asm volatile("global_load_async_to_lds_b128 %0, %1, %2"                  \
                   :: "v"(ldsB), "v"(goffB), "s"(X) : "memory");               \
    }                                                                          \
  } while (0)

  v8f acc[2][4] = {};               // [sub-row 0..1][sub-col 0..3], 16x16 each

  STAGE(0, 0);                      // prologue: fill buffer 0
  int buf = 0;

  for (int kb = 0; kb < DIM; kb += KT) {
    if (kb + KT < DIM) {
      STAGE(kb + KT, buf ^ 1);      // prefetch next slice into other buffer
      asm volatile("s_wait_asynccnt 0x4" ::: "memory");  // current slice landed
    } else {
      asm volatile("s_wait_asynccnt 0x0" ::: "memory");
    }
    __syncthreads();                // all waves' current slice visible

    // 8 x V_WMMA_F32_16X16X4_F32 per K=4 chunk, 4 chunks per slice
#pragma unroll
    for (int kc = 0; kc < KT; kc += 4) {
      // A fragment (16x4): lanes 0-15 hold K=kc,kc+1 ; lanes 16-31 K=kc+2,kc+3
      v2f a0 = *(const v2f*)&As[buf][(wrow * 32 +  0 + llo) * LSTR + kc + lhi * 2];
      v2f a1 = *(const v2f*)&As[buf][(wrow * 32 + 16 + llo) * LSTR + kc + lhi * 2];
      // B fragment (4x16): row n of the col band, same K pairing (B = X[J..]^T)
      v2f b0 = *(const v2f*)&Bs[buf][(wcol * 64 +  0 + llo) * LSTR + kc + lhi * 2];
      v2f b1 = *(const v2f*)&Bs[buf][(wcol * 64 + 16 + llo) * LSTR + kc + lhi * 2];
      v2f b2 = *(const v2f*)&Bs[buf][(wcol * 64 + 32 + llo) * LSTR + kc + lhi * 2];
      v2f b3 = *(const v2f*)&Bs[buf][(wcol * 64 + 48 + llo) * LSTR + kc + lhi * 2];

      acc[0][0] = __builtin_amdgcn_wmma_f32_16x16x4_f32(false, a0, false, b0, (short)0, acc[0][0], false, false);
      acc[0][1] = __builtin_amdgcn_wmma_f32_16x16x4_f32(false, a0, false, b1, (short)0, acc[0][1], false, false);
      acc[0][2] = __builtin_amdgcn_wmma_f32_16x16x4_f32(false, a0, false, b2, (short)0, acc[0][2], false, false);
      acc[0][3] = __builtin_amdgcn_wmma_f32_16x16x4_f32(false, a0, false, b3, (short)0, acc[0][3], false, false);
      acc[1][0] = __builtin_amdgcn_wmma_f32_16x16x4_f32(false, a1, false, b0, (short)0, acc[1][0], false, false);
      acc[1][1] = __builtin_amdgcn_wmma_f32_16x16x4_f32(false, a1, false, b1, (short)0, acc[1][1], false, false);
      acc[1][2] = __builtin_amdgcn_wmma_f32_16x16x4_f32(false, a1, false, b2, (short)0, acc[1][2], false, false);
      acc[1][3] = __builtin_amdgcn_wmma_f32_16x16x4_f32(false, a1, false, b3, (short)0, acc[1][3], false, false);
    }
    __syncthreads();                // done reading buf before it is refilled
    buf ^= 1;
  }
#undef STAGE

  // ---------------- masked argmin epilogue ----------------
  // C layout: VGPR r, lanes 0-15 -> M=r, N=lane ; lanes 16-31 -> M=8+r, N=lane-16
  const float BIG = 3.0e38f;
#pragma unroll
  for (int sr = 0; sr < 2; ++sr) {
#pragma unroll
    for (int r = 0; r < 8; ++r) {
      const int mloc = wrow * 32 + sr * 16 + lhi * 8 + r;
      const int grow = I + mloc;
      const int lr   = labR[mloc];

      float apv = BIG; int api = 0;
      float anv = BIG; int ani = 0;
#pragma unroll
      for (int sc = 0; sc < 4; ++sc) {      // ascending columns -> first-min ties
        const int nloc = wcol * 64 + sc * 16 + llo;
        const int gcol = J + nloc;
        const int lc   = labC[nloc];
        const float d  = acc[sr][sc][r];
        const bool same = (lr == lc);
        const bool eye  = (grow == gcol);
        const float dap = d + ((same && !eye) ? 0.0f : 2.0f);
        const float dan = d + ((!same)        ? 0.0f : 2.0f);
        if (dap < apv) { apv = dap; api = gcol; }
        if (dan < anv) { anv = dan; ani = gcol; }
      }
      // butterfly argmin across the 16 lanes of this half (stays inside half)
#pragma unroll
      for (int off = 1; off < 16; off <<= 1) {
        float ov = __shfl_xor(apv, off, 32);
        int   oi = __shfl_xor(api, off, 32);
        if (ov < apv || (ov == apv && oi < api)) { apv = ov; api = oi; }
        ov = __shfl_xor(anv, off, 32);
        oi = __shfl_xor(ani, off, 32);
        if (ov < anv || (ov == anv && oi < ani)) { anv = ov; ani = oi; }
      }
      if (llo == 0) {
        rvAP[wcol][mloc] = apv; riAP[wcol][mloc] = api;
        rvAN[wcol][mloc] = anv; riAN[wcol][mloc] = ani;
      }
    }
  }
  __syncthreads();

  // combine the two 64-col bands and emit the per-tile partial
  if (tid < TILE) {
    const int m = tid;
    float av = rvAP[0][m]; int ai = riAP[0][m];
    float bv = rvAP[1][m]; int bi = riAP[1][m];
    if (bv < av || (bv == av && bi < ai)) { av = bv; ai = bi; }

    float cv = rvAN[0][m]; int ci = riAN[0][m];
    float dv = rvAN[1][m]; int di = riAN[1][m];
    if (dv < cv || (dv == cv && di < ci)) { cv = dv; ci = di; }

    float4 p;
    p.x = av; p.y = __int_as_float(ai);
    p.z = cv; p.w = __int_as_float(ci);
    part[(size_t)(I + m) * NTIL + jt] = p;
  }
}

// -------------------------------------------------------------------------
// Phase 2: reduce the 32 per-col-tile partials per row, then gather
//          out[0:N*D]      = X[id_ap]
//          out[N*D:2*N*D]  = X[id_an]
// -------------------------------------------------------------------------
__global__ __launch_bounds__(256) void triplet_phase2(
    const float* __restrict__ X, const float4* __restrict__ part,
    float* __restrict__ out)
{
  const int row = blockIdx.x;
  __shared__ int sAP, sAN;

  if (threadIdx.x == 0) {
    float apv = 3.0e38f; int api = 0;
    float anv = 3.0e38f; int ani = 0;
    for (int t = 0; t < NTIL; ++t) {          // ascending -> first-min ties
      float4 p = part[(size_t)row * NTIL + t];
      int i0 = __float_as_int(p.y);
      int i1 = __float_as_int(p.w);
      if (p.x < apv || (p.x == apv && i0 < api)) { apv = p.x; api = i0; }
      if (p.z < anv || (p.z == anv && i1 < ani)) { anv = p.z; ani = i1; }
    }
    sAP = api; sAN = ani;
  }
  __syncthreads();

  const float4* srcA = (const float4*)(X + (size_t)sAP * DIM);
  const float4* srcB = (const float4*)(X + (size_t)sAN * DIM);
  float4* dstA = (float4*)(out + (size_t)row * DIM);
  float4* dstB = (float4*)(out + (size_t)(NROWS + row) * DIM);
  for (int c = threadIdx.x; c < DIM / 4; c += blockDim.x) {
    dstA[c] = srcA[c];
    dstB[c] = srcB[c];
  }
}

// -------------------------------------------------------------------------
extern "C" void kernel_launch(void* const* d_in, const int* in_sizes, int n_in,
                              void* d_out, int out_size, void* d_ws, size_t ws_size,
                              hipStream_t stream) {
  const float* X      = (const float*)d_in[0];   // [4096, 1024] fp32
  const int*   labels = (const int*)d_in[1];     // [4096] int
  float*       out    = (float*)d_out;           // [2, 4096, 1024] fp32
  float4*      part   = (float4*)d_ws;           // 4096 * 32 * 16B = 2 MB partials

  dim3 g1(NTIL, NTIL), b1(256);
  triplet_phase1<<<g1, b1, 0, stream>>>(X, labels, part);

  triplet_phase2<<<NROWS, 256, 0, stream>>>(X, part, out);
}